// NodeBlock_65249143161008
// MI455X (gfx1250) — compile-verified
//
#include <hip/hip_runtime.h>

#define N_NODES 50000
#define N_EDGES 800000
#define D       64
#define HID     256
#define KIN     128          // 2*D
#define MT      128          // rows of M per workgroup
#define PAD_A   132          // KIN + 4  -> bank-conflict-free stride
#define PAD_H   260          // HID + 4

typedef float v2f __attribute__((ext_vector_type(2)));
typedef float v8f __attribute__((ext_vector_type(8)));

// ---------------------------------------------------------------------------
// Kernel 1: zero the aggregation buffer (d_out doubles as agg scratch)
// ---------------------------------------------------------------------------
__global__ void zero_kernel(float4* __restrict__ out, int n4) {
    int i = blockIdx.x * blockDim.x + threadIdx.x;
    if (i < n4) out[i] = make_float4(0.f, 0.f, 0.f, 0.f);
}

// ---------------------------------------------------------------------------
// Kernel 2: scatter-sum edges into agg by sender AND receiver.
// One wave (32 lanes) per edge; lane p owns floats [2p, 2p+1] of the 64-wide
// edge feature -> one coalesced 256B row read + 4 hardware f32 atomics/lane.
// ---------------------------------------------------------------------------
__global__ __launch_bounds__(256) void scatter_kernel(
    const float* __restrict__ edges,
    const int*   __restrict__ senders,
    const int*   __restrict__ receivers,
    float*       __restrict__ agg) {
    long long t = (long long)blockIdx.x * blockDim.x + threadIdx.x;
    int e = (int)(t >> 5);
    if (e >= N_EDGES) return;
    int p = (int)(t & 31);

    int s = senders[e];
    int r = receivers[e];
    const float2 v = *(const float2*)(edges + (size_t)e * D + p * 2);

    float* ps = agg + (size_t)s * D + p * 2;
    float* pr = agg + (size_t)r * D + p * 2;
    unsafeAtomicAdd(ps + 0, v.x);
    unsafeAtomicAdd(ps + 1, v.y);
    unsafeAtomicAdd(pr + 0, v.x);
    unsafeAtomicAdd(pr + 1, v.y);
}

// ---------------------------------------------------------------------------
// Kernel 3: fused 2-layer MLP using V_WMMA_F32_16X16X4_F32.
// Per WG (256 threads = 8 waves): tile of MT=128 node rows.
//   sA[128][132] = [agg | nodes]  (64 KB+pad)  in LDS
//   sH[128][260] = relu(sA @ W1 + b1)          in LDS
//   out[128][64] = sH @ W2 + b2                -> d_out (overwrites agg rows)
// ---------------------------------------------------------------------------
__global__ __launch_bounds__(256, 1) void node_mlp_kernel(
    const float* __restrict__ nodes,
    const float* __restrict__ W1,
    const float* __restrict__ b1,
    const float* __restrict__ W2,
    const float* __restrict__ b2,
    float*       __restrict__ inout /* agg on entry, updated nodes on exit */) {

    __shared__ float sA[MT * PAD_A];
    __shared__ float sH[MT * PAD_H];

    const int tid  = threadIdx.x;
    const int wave = tid >> 5;
    const int lane = tid & 31;
    const int half = lane >> 4;   // 0: K lanes {0,1}, 1: K lanes {2,3}
    const int l16  = lane & 15;
    const int m0   = blockIdx.x * MT;

    // ---- Stage A: load collected = [agg | nodes] tile into LDS ------------
    for (int idx = tid; idx < MT * 32; idx += 256) {
        int row = idx >> 5;
        int q   = idx & 31;            // 32 float4's per 128-wide row
        int g   = m0 + row;
        float4 v = make_float4(0.f, 0.f, 0.f, 0.f);
        if (g < N_NODES) {
            if (q < 16) v = *(const float4*)(inout + (size_t)g * D + q * 4);
            else        v = *(const float4*)(nodes + (size_t)g * D + (q - 16) * 4);
        }
        *(float4*)(sA + row * PAD_A + q * 4) = v;
    }
    __syncthreads();

    // ---- GEMM1: h = relu(collected @ W1 + b1) -----------------------------
    // wave owns columns [wave*32, wave*32+32); two 16-wide N tiles.
    for (int nt = 0; nt < 2; ++nt) {
        const int n0 = wave * 32 + nt * 16 + l16;  // this lane's N column

        // Hoist the whole K=128 strip of B (W1) into registers: 32 x v2f.
        v2f Bv[32];
        #pragma unroll
        for (int ks = 0; ks < 32; ++ks) {
            const int k = ks * 4 + 2 * half;
            Bv[ks].x = W1[(size_t)(k + 0) * HID + n0];
            Bv[ks].y = W1[(size_t)(k + 1) * HID + n0];
        }
        const float bias = b1[n0];

        for (int mt = 0; mt < 8; ++mt) {           // 8 M-tiles of 16 rows
            // Prefetch the entire K=128 A-strip for this M-tile so the
            // 32-deep WMMA accumulation chain issues without dscnt stalls.
            const float* arow = sA + (mt * 16 + l16) * PAD_A + 2 * half;
            v2f Av[32];
            #pragma unroll
            for (int ks = 0; ks < 32; ++ks)
                Av[ks] = *(const v2f*)(arow + ks * 4);

            v8f acc;
            #pragma unroll
            for (int r = 0; r < 8; ++r) acc[r] = bias;

            #pragma unroll
            for (int ks = 0; ks < 32; ++ks)
                acc = __builtin_amdgcn_wmma_f32_16x16x4_f32(
                        false, Av[ks], false, Bv[ks], (short)0, acc,
                        false, false);

            // ReLU + stage into LDS (C layout: VGPR r -> row r / r+8)
            #pragma unroll
            for (int r = 0; r < 8; ++r) {
                float h = acc[r] > 0.f ? acc[r] : 0.f;
                sH[(mt * 16 + r + 8 * half) * PAD_H + n0] = h;
            }
        }
    }
    __syncthreads();

    // ---- GEMM2: out = h @ W2 + b2 ----------------------------------------
    // wave owns rows [wave*16, wave*16+16), all 64 output columns (4 N tiles).
    {
        v8f acc[4];
        #pragma unroll
        for (int nt = 0; nt < 4; ++nt) {
            const float bias = b2[nt * 16 + l16];
            #pragma unroll
            for (int r = 0; r < 8; ++r) acc[nt][r] = bias;
        }

        const float* arow = sH + (wave * 16 + l16) * PAD_H + 2 * half;
        for (int kc = 0; kc < 4; ++kc) {           // K chunks of 64
            v2f Av[16];
            #pragma unroll
            for (int kk = 0; kk < 16; ++kk)
                Av[kk] = *(const v2f*)(arow + kc * 64 + kk * 4);

            #pragma unroll
            for (int nt = 0; nt < 4; ++nt) {
                #pragma unroll
                for (int kk = 0; kk < 16; ++kk) {
                    const int k = kc * 64 + kk * 4 + 2 * half;
                    v2f b;
                    b.x = W2[(size_t)(k + 0) * D + nt * 16 + l16];
                    b.y = W2[(size_t)(k + 1) * D + nt * 16 + l16];
                    acc[nt] = __builtin_amdgcn_wmma_f32_16x16x4_f32(
                                false, Av[kk], false, b, (short)0, acc[nt],
                                false, false);
                }
            }
        }

        #pragma unroll
        for (int nt = 0; nt < 4; ++nt) {
            #pragma unroll
            for (int r = 0; r < 8; ++r) {
                const int grow = m0 + wave * 16 + r + 8 * half;
                if (grow < N_NODES)
                    inout[(size_t)grow * D + nt * 16 + l16] = acc[nt][r];
            }
        }
    }
}

// ---------------------------------------------------------------------------
extern "C" void kernel_launch(void* const* d_in, const int* in_sizes, int n_in,
                              void* d_out, int out_size, void* d_ws, size_t ws_size,
                              hipStream_t stream) {
    const float* nodes     = (const float*)d_in[0];
    const float* edges     = (const float*)d_in[1];
    const int*   senders   = (const int*)  d_in[2];
    const int*   receivers = (const int*)  d_in[3];
    const float* W1        = (const float*)d_in[4];
    const float* b1        = (const float*)d_in[5];
    const float* W2        = (const float*)d_in[6];
    const float* b2        = (const float*)d_in[7];
    float* out = (float*)d_out;

    // 1) zero agg (aliases d_out)
    const int n4 = (N_NODES * D) / 4;
    zero_kernel<<<(n4 + 255) / 256, 256, 0, stream>>>((float4*)out, n4);

    // 2) scatter-sum edges by senders + receivers
    const long long nthreads = (long long)N_EDGES * 32;
    scatter_kernel<<<(int)((nthreads + 255) / 256), 256, 0, stream>>>(
        edges, senders, receivers, out);

    // 3) fused 2-layer MLP (WMMA fp32), overwrites agg with final output
    node_mlp_kernel<<<(N_NODES + MT - 1) / MT, 256, 0, stream>>>(
        nodes, W1, b1, W2, b2, out);
}